// HostGraphSAGE_31714038513706
// MI455X (gfx1250) — compile-verified
//
#include <hip/hip_runtime.h>
#include <math.h>

typedef __attribute__((ext_vector_type(2))) float v2f;
typedef __attribute__((ext_vector_type(8))) float v8f;

#define NNODES 50000
#define NEDGES 800000

// D = A(16x4 f32) * B(4x16 f32) + C(16x16 f32), full-precision WMMA (CDNA5)
__device__ __forceinline__ v8f wmma4(v2f a, v2f b, v8f c) {
  return __builtin_amdgcn_wmma_f32_16x16x4_f32(
      /*neg_a=*/false, a, /*neg_b=*/false, b,
      /*c_mod=*/(short)0, c, /*reuse_a=*/false, /*reuse_b=*/false);
}

__global__ void zero_kernel(float* __restrict__ p, long long n) {
  long long i = (long long)blockIdx.x * blockDim.x + threadIdx.x;
  long long stride = (long long)gridDim.x * blockDim.x;
  for (; i < n; i += stride) p[i] = 0.0f;
}

__global__ void degree_kernel(const int* __restrict__ ei, float* __restrict__ deg, int E) {
  int e = blockIdx.x * blockDim.x + threadIdx.x;
  if (e < E) atomicAdd(&deg[ei[E + e]], 1.0f);
}

// One float4 feature group per thread per edge; atomics land in L2 (agg fits in 192MB L2).
template <int D>
__global__ void scatter_add_kernel(const float* __restrict__ xin, const int* __restrict__ ei,
                                   float* __restrict__ agg, int E) {
  constexpr int G = D / 4;      // float4 groups per node row
  constexpr int EPB = 256 / G;  // edges per 256-thread block
  const int t = threadIdx.x;
  const int e = blockIdx.x * EPB + t / G;
  if (e >= E) return;
  const int fg = t % G;
  const int s = ei[e];
  const int d = ei[E + e];
  const float4 v = *reinterpret_cast<const float4*>(xin + (long long)s * D + fg * 4);
  float* dp = agg + (long long)d * D + fg * 4;
  atomicAdd(dp + 0, v.x);
  atomicAdd(dp + 1, v.y);
  atomicAdd(dp + 2, v.z);
  atomicAdd(dp + 3, v.w);
}

// out[row, col] = relu?(BN( (agg/deg)@Wl + bl + xin@Wr ))
// One wave computes a 16x16 tile with v_wmma_f32_16x16x4_f32 (2 WMMAs per K-step).
// A layout (ISA 7.12.2, 32-bit A 16x4): lanes 0-15 row M=lane, K=0(v0)/1(v1);
//                                       lanes 16-31 row M=lane-16, K=2(v0)/3(v1).
// B assumed symmetric: lane half selects K pair, lane&15 selects column N.
// C/D: VGPR r -> M = r + 8*(lane>>4), N = lane&15.
template <int D_IN, bool BN_RELU, bool MASK>
__global__ void sage_gemm_kernel(const float* __restrict__ agg, const float* __restrict__ xin,
                                 const float* __restrict__ deg,
                                 const float* __restrict__ Wl, const float* __restrict__ bl,
                                 const float* __restrict__ Wr,
                                 const float* __restrict__ bn_g, const float* __restrict__ bn_b,
                                 const float* __restrict__ bn_m, const float* __restrict__ bn_v,
                                 float* __restrict__ out, int nvalid, int ld_out) {
  const int wave = threadIdx.x >> 5;
  const int lane = threadIdx.x & 31;
  const int hf   = lane >> 4;   // which K-pair / which M-half for C/D
  const int mr   = lane & 15;
  const int wavesPerBlock = blockDim.x >> 5;
  const int col0 = (blockIdx.y * wavesPerBlock + wave) * 16;
  const int row0 = blockIdx.x * 16;

  const float invdeg = 1.0f / fmaxf(deg[row0 + mr], 1.0f);
  const int cb = col0 + mr;   // this lane's output / B column
  const int ldw = nvalid;     // weights are row-major [D_IN, nvalid]
  const bool colok = MASK ? (cb < nvalid) : true;

  const float* arow = agg + (long long)(row0 + mr) * D_IN;
  const float* xrow = xin + (long long)(row0 + mr) * D_IN;

  v8f acc = {};
#pragma unroll 4
  for (int k = 0; k < D_IN; k += 4) {
    const int ka = k + 2 * hf;
    v2f aAgg, aX, bL, bR;
    aAgg.x = arow[ka] * invdeg;
    aAgg.y = arow[ka + 1] * invdeg;
    aX.x = xrow[ka];
    aX.y = xrow[ka + 1];
    if (MASK) {
      bL.x = colok ? Wl[(long long)ka * ldw + cb] : 0.0f;
      bL.y = colok ? Wl[(long long)(ka + 1) * ldw + cb] : 0.0f;
      bR.x = colok ? Wr[(long long)ka * ldw + cb] : 0.0f;
      bR.y = colok ? Wr[(long long)(ka + 1) * ldw + cb] : 0.0f;
    } else {
      bL.x = Wl[(long long)ka * ldw + cb];
      bL.y = Wl[(long long)(ka + 1) * ldw + cb];
      bR.x = Wr[(long long)ka * ldw + cb];
      bR.y = Wr[(long long)(ka + 1) * ldw + cb];
    }
    acc = wmma4(aAgg, bL, acc);   // (agg/deg) @ Wl
    acc = wmma4(aX, bR, acc);     // x @ Wr
  }

  float bias = 0.0f, scale = 1.0f, shift = 0.0f;
  if (colok) {
    bias = bl[cb];
    if (BN_RELU) {
      scale = bn_g[cb] * rsqrtf(bn_v[cb] + 1e-5f);
      shift = bn_b[cb] - bn_m[cb] * scale;
    }
  }
#pragma unroll
  for (int r = 0; r < 8; ++r) {
    const int row = row0 + r + 8 * hf;
    float val = acc[r] + bias;
    if (BN_RELU) val = fmaxf(val * scale + shift, 0.0f);
    if (colok) out[(long long)row * ld_out + cb] = val;
  }
}

// h2 is [N,16] padded (10 valid); out is [N,10] log_softmax.
__global__ void log_softmax_kernel(const float* __restrict__ h2, float* __restrict__ out, int n) {
  int i = blockIdx.x * blockDim.x + threadIdx.x;
  if (i >= n) return;
  const float* r = h2 + (long long)i * 16;
  float mx = r[0];
#pragma unroll
  for (int j = 1; j < 10; ++j) mx = fmaxf(mx, r[j]);
  float s = 0.0f;
#pragma unroll
  for (int j = 0; j < 10; ++j) s += __expf(r[j] - mx);
  float lse = mx + __logf(s);
#pragma unroll
  for (int j = 0; j < 10; ++j) out[(long long)i * 10 + j] = r[j] - lse;
}

extern "C" void kernel_launch(void* const* d_in, const int* in_sizes, int n_in,
                              void* d_out, int out_size, void* d_ws, size_t ws_size,
                              hipStream_t stream) {
  const float* x   = (const float*)d_in[0];
  const int*   ei  = (const int*)d_in[1];
  const float* Wl0 = (const float*)d_in[2];
  const float* bl0 = (const float*)d_in[3];
  const float* Wr0 = (const float*)d_in[4];
  const float* g0  = (const float*)d_in[5];
  const float* b0  = (const float*)d_in[6];
  const float* m0  = (const float*)d_in[7];
  const float* v0  = (const float*)d_in[8];
  const float* Wl1 = (const float*)d_in[9];
  const float* bl1 = (const float*)d_in[10];
  const float* Wr1 = (const float*)d_in[11];
  const float* g1  = (const float*)d_in[12];
  const float* b1  = (const float*)d_in[13];
  const float* m1  = (const float*)d_in[14];
  const float* v1  = (const float*)d_in[15];
  const float* Wl2 = (const float*)d_in[16];
  const float* bl2 = (const float*)d_in[17];
  const float* Wr2 = (const float*)d_in[18];
  float* out = (float*)d_out;

  float* ws  = (float*)d_ws;
  float* deg = ws;                               // N floats
  float* agg = deg + NNODES;                     // N*128 floats
  float* hA  = agg + (size_t)NNODES * 128;       // N*128 floats
  float* hB  = hA  + (size_t)NNODES * 128;       // N*128 floats

  const int E = NEDGES, Nn = NNODES;

  // degree (shared by all layers) + first agg buffer, zero-init
  zero_kernel<<<2048, 256, 0, stream>>>(deg, (long long)Nn * 129);  // deg + agg contiguous
  degree_kernel<<<(E + 255) / 256, 256, 0, stream>>>(ei, deg, E);

  // ---- layer 0: d_in=64 -> d_out=128, BN+ReLU ----
  scatter_add_kernel<64><<<(E + 15) / 16, 256, 0, stream>>>(x, ei, agg, E);
  {
    dim3 grid(Nn / 16, 2);  // 2 blocks of 4 waves cover 128 output cols
    sage_gemm_kernel<64, true, false><<<grid, 128, 0, stream>>>(
        agg, x, deg, Wl0, bl0, Wr0, g0, b0, m0, v0, hA, 128, 128);
  }

  // ---- layer 1: 128 -> 128, BN+ReLU ----
  zero_kernel<<<2048, 256, 0, stream>>>(agg, (long long)Nn * 128);
  scatter_add_kernel<128><<<(E + 7) / 8, 256, 0, stream>>>(hA, ei, agg, E);
  {
    dim3 grid(Nn / 16, 2);
    sage_gemm_kernel<128, true, false><<<grid, 128, 0, stream>>>(
        agg, hA, deg, Wl1, bl1, Wr1, g1, b1, m1, v1, hB, 128, 128);
  }

  // ---- layer 2: 128 -> 10 (padded to 16), no BN/ReLU ----
  zero_kernel<<<2048, 256, 0, stream>>>(agg, (long long)Nn * 128);
  scatter_add_kernel<128><<<(E + 7) / 8, 256, 0, stream>>>(hB, ei, agg, E);
  {
    dim3 grid(Nn / 16, 1);
    sage_gemm_kernel<128, false, true><<<grid, 32, 0, stream>>>(
        agg, hB, deg, Wl2, bl2, Wr2, nullptr, nullptr, nullptr, nullptr, hA, 10, 16);
  }

  // ---- log_softmax over 10 classes ----
  log_softmax_kernel<<<(Nn + 255) / 256, 256, 0, stream>>>(hA, out, Nn);
}